// My_loss_36275293782593
// MI455X (gfx1250) — compile-verified
//
#include <hip/hip_runtime.h>
#include <math.h>

typedef __attribute__((ext_vector_type(2))) float v2f;
typedef __attribute__((ext_vector_type(8))) float v8f;

#define THREADS 256
#define ELEMS   4

// ---------------------------------------------------------------------------
// Kernel 1: streaming HMM forward scan. One thread = one batch element at a
// time (ELEMS elements per thread). Per-thread emission LUT lives in LDS:
// slot o (0..5) holds (E0[o], E1[o], E2[o], 0); slot 6 is zero (NaN sink).
// Per-block partial of sum(-log(sum(alpha))) written to d_ws (deterministic).
// ---------------------------------------------------------------------------
__global__ __launch_bounds__(THREADS) void hmm_fwd_partial(
    const float* __restrict__ Y,
    const float* __restrict__ Yo,
    float* __restrict__ partials,
    int B)
{
    __shared__ float4 tbl[THREADS * 7];
    __shared__ float  red[THREADS];

    const int tid = threadIdx.x;
    float4* __restrict__ my = &tbl[tid * 7];

    // Global scalar from the reference's quirk: 1 - Y[batch=1, col=3]
    const float c15 = 1.0f - Y[16];

    float acc = 0.0f;
    const int base = blockIdx.x * (THREADS * ELEMS);

    for (int e = 0; e < ELEMS; ++e) {
        const int b = base + e * THREADS + tid;
        if (b < B) {
            const float* __restrict__ y = Y + (size_t)b * 13;
            const float y0 = y[0], y1 = y[1], y2 = y[2], y3 = y[3];
            const float y4 = y[4], y5 = y[5], y6 = y[6];
            const float t00 = y[7],  t01 = y[8];
            const float t10 = y[9],  t11 = y[10];
            const float t20 = y[11], t21 = y[12];
            const float t02 = 1.0f - t00 - t01;
            const float t12 = 1.0f - t10 - t11;
            const float t22 = 1.0f - t20 - t21;

            // Emission LUT (columns of E), per-thread private LDS region.
            my[0] = make_float4(y0, 0.0f, y4, 0.0f);
            my[1] = make_float4(y1, 0.0f, y5, 0.0f);
            my[2] = make_float4(y2, 0.0f, y6, 0.0f);
            my[3] = make_float4(1.0f - y0 - y1 - y2, 0.0f,
                                1.0f - y4 - y5 - y6, 0.0f);
            my[4] = make_float4(0.0f, y3,  0.0f, 0.0f);
            my[5] = make_float4(0.0f, c15, 0.0f, 0.0f);
            my[6] = make_float4(0.0f, 0.0f, 0.0f, 0.0f);  // NaN sink

            // Observation row: 32 floats = 8 x float4 (rows are 128B aligned).
            const float4* __restrict__ yo = (const float4*)(Yo + (size_t)b * 32);
            float o[32];
            #pragma unroll
            for (int q = 0; q < 8; ++q) {
                const float4 v = yo[q];
                o[4*q+0] = v.x; o[4*q+1] = v.y; o[4*q+2] = v.z; o[4*q+3] = v.w;
            }

            // alpha init: Pi_0 (ones) * E[:, obs0]   (obs0 always valid: len>=4)
            const float4 e0 = my[(int)o[0]];
            float a0 = e0.x, a1 = e0.y, a2 = e0.z;

            // Forward scan; mask == !isnan(obs_t) (NaN fill is a suffix).
            #pragma unroll
            for (int t = 1; t < 32; ++t) {
                const float v     = o[t];
                const bool  valid = (v == v);
                const float vs    = valid ? v : 6.0f;
                const float4 em   = my[(int)vs];
                const float n0 = (a0 * t00 + a1 * t10 + a2 * t20) * em.x;
                const float n1 = (a0 * t01 + a1 * t11 + a2 * t21) * em.y;
                const float n2 = (a0 * t02 + a1 * t12 + a2 * t22) * em.z;
                a0 = valid ? n0 : a0;
                a1 = valid ? n1 : a1;
                a2 = valid ? n2 : a2;
            }
            acc -= logf(a0 + a1 + a2);
        }
    }

    // Deterministic block reduction (fixed tree order).
    red[tid] = acc;
    __syncthreads();
    for (int s = THREADS / 2; s > 0; s >>= 1) {
        if (tid < s) red[tid] += red[tid + s];
        __syncthreads();
    }
    if (tid == 0) partials[blockIdx.x] = red[0];
}

// ---------------------------------------------------------------------------
// Kernel 2: single-wave deterministic reduction of block partials using
// V_WMMA_F32_16X16X4_F32 with an all-ones A matrix. Every row of D equals the
// per-column sums of B, so summing D row 0 across its 16 lanes yields the
// exact total of each 64-value chunk regardless of B's lane layout.
// ---------------------------------------------------------------------------
__global__ __launch_bounds__(32) void reduce_wmma(
    const float* __restrict__ p, float* __restrict__ out, int n, float invB)
{
    const int lane = threadIdx.x;
    v8f c = {};
    v2f a; a.x = 1.0f; a.y = 1.0f;  // A = ones(16x4), layout-independent

    for (int base = 0; base < n; base += 64) {
        v2f bv;
        const int i0 = base + lane;
        const int i1 = base + 32 + lane;
        bv.x = (i0 < n) ? p[i0] : 0.0f;
        bv.y = (i1 < n) ? p[i1] : 0.0f;
        // 8 args: (neg_a, A, neg_b, B, c_mod, C, reuse_a, reuse_b)
        c = __builtin_amdgcn_wmma_f32_16x16x4_f32(
                false, a, false, bv, (short)0, c, false, false);
    }

    // D row M=0 lives in c[0] on lanes 0..15 (N = lane). Sum those 16 lanes.
    float r = c[0];
    r += __shfl_xor(r, 1, 32);
    r += __shfl_xor(r, 2, 32);
    r += __shfl_xor(r, 4, 32);
    r += __shfl_xor(r, 8, 32);
    if (lane == 0) out[0] = r * invB;
}

// ---------------------------------------------------------------------------
extern "C" void kernel_launch(void* const* d_in, const int* in_sizes, int n_in,
                              void* d_out, int out_size, void* d_ws, size_t ws_size,
                              hipStream_t stream)
{
    const float* Y  = (const float*)d_in[0];  // (B, 13) f32
    const float* Yo = (const float*)d_in[1];  // (B, 32) f32 (NaN-masked obs)
    // d_in[2] = Pi_0 (ones, int32) -> contributes factor 1.0, unused.

    const int B    = in_sizes[0] / 13;
    const int nblk = (B + THREADS * ELEMS - 1) / (THREADS * ELEMS);

    float* partials = (float*)d_ws;  // nblk floats (every slot rewritten below)

    hmm_fwd_partial<<<nblk, THREADS, 0, stream>>>(Y, Yo, partials, B);
    reduce_wmma<<<1, 32, 0, stream>>>(partials, (float*)d_out, nblk,
                                      1.0f / (float)B);
}